// nu_grid_sampler_simple_37890201485783
// MI455X (gfx1250) — compile-verified
//
#include <hip/hip_runtime.h>
#include <hip/hip_bf16.h>
#include <stdint.h>

// Problem constants (from reference): x (B,C,NX,NY) f32, coords (B,N,2) f32,
// out (B,C,N) f32.
#define B_  8
#define C_  128
#define NX_ 256
#define NY_ 256
#define N_  32768
#define PLANE_ELEMS (NX_ * NY_)          // 65536 floats
#define PLANE_BYTES (PLANE_ELEMS * 4)    // 262144 bytes = 256 KiB (fits 320 KB LDS)
#define TPB 256                          // 8 wave32s
#define COPY_ROUNDS (PLANE_BYTES / (TPB * 16))   // 64 rounds of b128 per thread

typedef float v2f __attribute__((ext_vector_type(2)));
typedef float v4f __attribute__((ext_vector_type(4)));

// Issue COPY_ROUNDS async global->LDS b128 copies that differ only in the
// signed-24-bit instruction offset (applied to BOTH the LDS and global
// address per the CDNA5 ISA), so there is zero per-round address math.
template <int R>
__device__ __forceinline__ void async_plane_copy(unsigned loff,
                                                 unsigned long long ga) {
    asm volatile("global_load_async_to_lds_b128 %0, %1, off offset:%c2"
                 :: "v"(loff), "v"(ga), "i"(R * (TPB * 16)) : "memory");
    if constexpr (R + 1 < COPY_ROUNDS) async_plane_copy<R + 1>(loff, ga);
}

// Reference index math: clip(v*(dim-1), 0, dim) -> int32 (trunc); the jnp
// gather then clips the flat index, replicated by the final min().
__device__ __forceinline__ int flat_index(float c0, float c1) {
    const float fy = fminf(fmaxf(c0 * (float)(NY_ - 1), 0.0f), (float)NY_);
    const float fx = fminf(fmaxf(c1 * (float)(NX_ - 1), 0.0f), (float)NX_);
    int f = (int)fx * NY_ + (int)fy;
    return f < (PLANE_ELEMS - 1) ? f : (PLANE_ELEMS - 1);
}

// One workgroup per (b, c) plane:
//   1) async-copy the 256 KiB channel plane global -> LDS (ASYNCcnt path)
//   2) gather N samples from LDS (2 samples / thread / iter)
//   3) non-temporal coalesced b64 stores of the output row
__global__ __launch_bounds__(TPB, 1)
void nu_grid_sampler_gather_kernel(const float* __restrict__ x,
                                   const float* __restrict__ coords,
                                   float* __restrict__ out) {
    extern __shared__ float smem[];  // PLANE_BYTES dynamic LDS

    const int bid = blockIdx.x;          // 0 .. B_*C_-1
    const int b   = bid >> 7;            // bid / C_
    const int tid = threadIdx.x;

    // ---- Phase 1: stream plane into LDS. Per-thread base, offsets do the rest.
    const unsigned long long gbase =
        (unsigned long long)(x + (size_t)bid * PLANE_ELEMS) +
        (unsigned long long)tid * 16ull;
    // Low 32 bits of a generic LDS pointer are the LDS byte address.
    const unsigned lbase = (unsigned)(size_t)smem + (unsigned)tid * 16u;
    async_plane_copy<0>(lbase, gbase);

    // Wait for this wave's async copies, then sync all 8 waves of the block.
    asm volatile("s_wait_asynccnt 0x0" ::: "memory");
    __syncthreads();

    // ---- Phase 2: gather from LDS, two samples per thread per iteration.
    // coords[b, n, 0] -> py (ny axis), coords[b, n, 1] -> px (nx axis).
    const v4f* __restrict__ c4 =
        (const v4f*)(coords + (size_t)b * N_ * 2);   // c4[i] = points 2i, 2i+1
    v2f* __restrict__ o2 = (v2f*)(out + (size_t)bid * N_);

    #pragma unroll 4
    for (int i = tid; i < N_ / 2; i += TPB) {
        const v4f cc = c4[i];                        // b128 coalesced load
        const int f0 = flat_index(cc.x, cc.y);
        const int f1 = flat_index(cc.z, cc.w);
        v2f v;
        v.x = smem[f0];                              // ds_load_b32 gather
        v.y = smem[f1];
        __builtin_nontemporal_store(v, &o2[i]);      // streamed b64 store
    }
}

extern "C" void kernel_launch(void* const* d_in, const int* in_sizes, int n_in,
                              void* d_out, int out_size, void* d_ws, size_t ws_size,
                              hipStream_t stream) {
    (void)in_sizes; (void)n_in; (void)d_ws; (void)ws_size; (void)out_size;
    const float* x      = (const float*)d_in[0];   // (B, C, NX, NY) f32
    const float* coords = (const float*)d_in[1];   // (B, N, 2) f32
    float*       out    = (float*)d_out;           // (B, C, N) f32

    dim3 grid(B_ * C_);   // 1024 blocks, one 256 KiB plane each
    dim3 block(TPB);
    nu_grid_sampler_gather_kernel<<<grid, block, PLANE_BYTES, stream>>>(x, coords, out);
}